// SCANN_82111184764987
// MI455X (gfx1250) — compile-verified
//
#include <hip/hip_runtime.h>
#include <hip/hip_bf16.h>

typedef __attribute__((ext_vector_type(2))) float v2f;
typedef __attribute__((ext_vector_type(8))) float v8f;

#define BG    1024        // B*G = 16*64
#define GRP   64          // G (channels per batch)
#define HH    224
#define WW    224
#define PSIDE 112
#define NACC  54          // 9 (m) + 45 (upper-tri of 9x9 Gram)
#define PF    12544.0f    // number of patches

// triangular index for (i<=j): 9 + 9*i - i*(i-1)/2 + (j-i)
__device__ __forceinline__ int triIdx(int i, int j) {
    return 9 + 9 * i - (i * (i - 1)) / 2 + (j - i);
}

// ---------------------------------------------------------------------------
// Kernel 1: per-(b,g) Gram reduction + WMMA energy chain + softmax epilogue.
// One block of 256 threads per (b,g). Outputs coef[gid*16 + 0..8] = c, [9] = d.
// ---------------------------------------------------------------------------
__global__ __launch_bounds__(256)
void scann_stats_kernel(const float* __restrict__ x,
                        const float* __restrict__ Wq, const float* __restrict__ bq,
                        const float* __restrict__ Wk, const float* __restrict__ bk,
                        const float* __restrict__ Wv, const float* __restrict__ bv,
                        const float* __restrict__ Wo, const float* __restrict__ bo,
                        float* __restrict__ coef)
{
    const int gid = blockIdx.x;          // b*G + g
    const int g   = gid & (GRP - 1);
    const int tid = threadIdx.x;
    const float* __restrict__ xc = x + (size_t)gid * (HH * WW);

    __shared__ float red[8 * NACC];      // per-wave partials
    __shared__ float ssum[NACC];         // reduced stats: [0..8]=m, [9..53]=triM
    __shared__ float Mp[256], Qp[256], Kp[256], T1[256], E1[256];
    __shared__ float qmv[9], kmv[9], Emat[81], wrow[81], weff[9];

    float acc[NACC];
#pragma unroll
    for (int i = 0; i < NACC; ++i) acc[i] = 0.0f;

    // --- Phase A: sliding-window Gram accumulation (224 workers) ---
    if (tid < 224) {
        const int ph    = tid >> 1;
        const int pw0   = (tid & 1) * 56;
        const int r0    = 2 * ph - 1;          // top window row (may be -1)
        const bool topok = (ph > 0);

        // preload column at image col (2*pw0 - 1)
        float a0, a1, a2;
        {
            const int c = 2 * pw0 - 1;
            const bool cok = (c >= 0);
            a0 = (cok && topok) ? xc[r0 * WW + c]       : 0.0f;
            a1 = cok            ? xc[(r0 + 1) * WW + c] : 0.0f;
            a2 = cok            ? xc[(r0 + 2) * WW + c] : 0.0f;
        }
        for (int pw = pw0; pw < pw0 + 56; ++pw) {
            const int c1 = 2 * pw, c2 = 2 * pw + 1;   // always within [0,224)
            const float b0 = topok ? xc[r0 * WW + c1] : 0.0f;
            const float b1 = xc[(r0 + 1) * WW + c1];
            const float b2 = xc[(r0 + 2) * WW + c1];
            const float d0 = topok ? xc[r0 * WW + c2] : 0.0f;
            const float d1 = xc[(r0 + 1) * WW + c2];
            const float d2 = xc[(r0 + 2) * WW + c2];
            const float v[9] = { a0, b0, d0,  a1, b1, d1,  a2, b2, d2 };
#pragma unroll
            for (int i = 0; i < 9; ++i) acc[i] += v[i];
#pragma unroll
            for (int i = 0; i < 9; ++i) {
#pragma unroll
                for (int j = i; j < 9; ++j)
                    acc[triIdx(i, j)] += v[i] * v[j];
            }
            a0 = d0; a1 = d1; a2 = d2;   // stride-2: shared column becomes col0
        }
    }

    // --- Phase B: deterministic wave32 shuffle reduction, then cross-wave ---
#pragma unroll
    for (int k = 0; k < NACC; ++k) {
        float v = acc[k];
#pragma unroll
        for (int off = 16; off > 0; off >>= 1) v += __shfl_xor(v, off, 32);
        acc[k] = v;
    }
    const int wid = tid >> 5;
    if ((tid & 31) == 0) {
#pragma unroll
        for (int k = 0; k < NACC; ++k) red[wid * NACC + k] = acc[k];
    }
    __syncthreads();
    if (tid < 32) {
        for (int k = tid; k < NACC; k += 32) {
            float s = 0.0f;
#pragma unroll
            for (int w = 0; w < 8; ++w) s += red[w * NACC + k];
            ssum[k] = s;
        }
    }
    __syncthreads();

    // --- Phase C: build zero-padded 16x16 operands (all 256 threads) ---
    {
        const int i = tid >> 4, j = tid & 15;
        float mv = 0.0f, qv = 0.0f, kv = 0.0f;
        if (i < 9 && j < 9) {
            const int ii = (i < j) ? i : j;
            const int jj = (i < j) ? j : i;
            mv = ssum[triIdx(ii, jj)];
            qv = Wq[g * 81 + i * 9 + j];
            kv = Wk[g * 81 + i * 9 + j];
        }
        Mp[tid] = mv; Qp[tid] = qv; Kp[tid] = kv;
    }
    __syncthreads();

    // --- Phase D: E1 = Wk * M * Wq^T via V_WMMA_F32_16X16X4_F32 (wave 0) ---
    // Step 1: T1 = M @ Wq^T   (A = M chunk, B[k,col] = Wq[col,k])
    if (tid < 32) {
        const int row = tid & 15, grp = tid >> 4;  // col == row for B/D access
        v8f t1 = {};
#pragma unroll
        for (int jc = 0; jc < 3; ++jc) {
            const int kb = 4 * jc + 2 * grp;
            v2f a, b;
            a.x = Mp[row * 16 + kb]; a.y = Mp[row * 16 + kb + 1];
            b.x = Qp[row * 16 + kb]; b.y = Qp[row * 16 + kb + 1];
            t1 = __builtin_amdgcn_wmma_f32_16x16x4_f32(false, a, false, b,
                                                       (short)0, t1, false, false);
        }
#pragma unroll
        for (int v = 0; v < 8; ++v) T1[(v + 8 * grp) * 16 + row] = t1[v];
    }
    __syncthreads();
    // Step 2: E1 = Wk @ T1
    if (tid < 32) {
        const int row = tid & 15, grp = tid >> 4;
        v8f e = {};
#pragma unroll
        for (int jc = 0; jc < 3; ++jc) {
            const int kb = 4 * jc + 2 * grp;
            v2f a, b;
            a.x = Kp[row * 16 + kb];      a.y = Kp[row * 16 + kb + 1];
            b.x = T1[kb * 16 + row];      b.y = T1[(kb + 1) * 16 + row];
            e = __builtin_amdgcn_wmma_f32_16x16x4_f32(false, a, false, b,
                                                      (short)0, e, false, false);
        }
#pragma unroll
        for (int v = 0; v < 8; ++v) E1[(v + 8 * grp) * 16 + row] = e[v];
    }
    __syncthreads();

    // --- Phase E: bias/rank-1 terms, softmax, collapse to conv coeffs ---
    if (tid < 18) {
        const int t = (tid < 9) ? tid : tid - 9;
        const float* Wrow = (tid < 9) ? &Qp[t * 16] : &Kp[t * 16];
        float s = 0.0f;
#pragma unroll
        for (int a = 0; a < 9; ++a) s += Wrow[a] * ssum[a];
        if (tid < 9) qmv[t] = s; else kmv[t] = s;
    }
    __syncthreads();
    if (tid < 81) {
        const int s_ = tid / 9, t_ = tid % 9;
        const float bks = bk[g * 9 + s_], bqt = bq[g * 9 + t_];
        Emat[tid] = E1[s_ * 16 + t_] + bks * qmv[t_] + bqt * kmv[s_] + PF * bks * bqt;
    }
    __syncthreads();
    if (tid < 9) {   // row-wise softmax scaled by Wo
        const int s_ = tid;
        float mx = -3.0e38f;
#pragma unroll
        for (int t = 0; t < 9; ++t) mx = fmaxf(mx, Emat[s_ * 9 + t]);
        float ex[9], sum = 0.0f;
#pragma unroll
        for (int t = 0; t < 9; ++t) { ex[t] = __expf(Emat[s_ * 9 + t] - mx); sum += ex[t]; }
        const float wo = Wo[g * 9 + s_] / sum;
#pragma unroll
        for (int t = 0; t < 9; ++t) wrow[s_ * 9 + t] = wo * ex[t];
    }
    __syncthreads();
    if (tid < 9) {
        float s = 0.0f;
#pragma unroll
        for (int r = 0; r < 9; ++r) s += wrow[r * 9 + tid];
        weff[tid] = s;
    }
    __syncthreads();
    if (tid < 10) {
        if (tid < 9) {
            float s = 0.0f;
#pragma unroll
            for (int t = 0; t < 9; ++t) s += weff[t] * Wv[g * 81 + t * 9 + tid];
            coef[gid * 16 + tid] = s;
        } else {
            float s = bo[g];
#pragma unroll
            for (int t = 0; t < 9; ++t) s += weff[t] * bv[g * 9 + t];
            coef[gid * 16 + 9] = s;
        }
    }
}

// ---------------------------------------------------------------------------
// Kernel 2: dynamic per-(b,g) 3x3 stride-2 conv. 16x16 output tile per block,
// 33x33 input region staged through LDS.
// ---------------------------------------------------------------------------
__global__ __launch_bounds__(256)
void scann_apply_kernel(const float* __restrict__ x,
                        const float* __restrict__ coef,
                        float* __restrict__ out)
{
    const int gid = blockIdx.z;
    const int pw0 = blockIdx.x * 16;
    const int ph0 = blockIdx.y * 16;
    const int tid = threadIdx.x;

    __shared__ float tile[33 * 34];
    __shared__ float cc[10];

    if (tid < 10) cc[tid] = coef[gid * 16 + tid];

    const float* __restrict__ xc = x + (size_t)gid * (HH * WW);
    const int rb = 2 * ph0 - 1, cb = 2 * pw0 - 1;
    for (int i = tid; i < 33 * 33; i += 256) {
        const int rr = i / 33, cl = i % 33;
        const int r = rb + rr, c = cb + cl;
        float v = 0.0f;
        if ((unsigned)r < (unsigned)HH && (unsigned)c < (unsigned)WW)
            v = xc[r * WW + c];
        tile[rr * 34 + cl] = v;
    }
    __syncthreads();

    const int ty = tid >> 4, tx = tid & 15;
    float s = cc[9];
#pragma unroll
    for (int kh = 0; kh < 3; ++kh) {
#pragma unroll
        for (int kw = 0; kw < 3; ++kw)
            s += cc[kh * 3 + kw] * tile[(2 * ty + kh) * 34 + (2 * tx + kw)];
    }
    out[((size_t)gid * PSIDE + (ph0 + ty)) * PSIDE + (pw0 + tx)] = s;
}

// ---------------------------------------------------------------------------
extern "C" void kernel_launch(void* const* d_in, const int* in_sizes, int n_in,
                              void* d_out, int out_size, void* d_ws, size_t ws_size,
                              hipStream_t stream)
{
    const float* x  = (const float*)d_in[0];
    const float* Wq = (const float*)d_in[1];
    const float* bq = (const float*)d_in[2];
    const float* Wk = (const float*)d_in[3];
    const float* bk = (const float*)d_in[4];
    const float* Wv = (const float*)d_in[5];
    const float* bv = (const float*)d_in[6];
    const float* Wo = (const float*)d_in[7];
    const float* bo = (const float*)d_in[8];
    float* out  = (float*)d_out;
    float* coef = (float*)d_ws;            // BG * 16 floats = 64 KB

    scann_stats_kernel<<<BG, 256, 0, stream>>>(x, Wq, bq, Wk, bk, Wv, bv, Wo, bo, coef);
    scann_apply_kernel<<<dim3(7, 7, BG), dim3(256, 1, 1), 0, stream>>>(x, coef, out);
}